// Signature_41824391528728
// MI455X (gfx1250) — compile-verified
//
#include <hip/hip_runtime.h>

typedef float v2f __attribute__((ext_vector_type(2)));
typedef float v8f __attribute__((ext_vector_type(8)));

#define N_SAMPLES 32
#define LPTS 128
#define C 48
#define TSTEPS 127                 // real increments
#define NCHUNK 32                  // padded T = 128 = 32 chunks of K=4
#define C2 (C * C)                 // 2304
#define OUT_STRIDE (C + C2 + C2 * C)   // 112944
#define S3_OFF (C + C2)            // 2352
#define BLOCKS_PER_SAMPLE 54       // 432 tiles / 8 waves

__global__ __launch_bounds__(256)
void sig3_fused_kernel(const float* __restrict__ path, float* __restrict__ out)
{
    // double-buffered increment chunks: [buf][tt in chunk][channel]
    __shared__ float vbuf[2][4][C];

    const int n    = blockIdx.x / BLOCKS_PER_SAMPLE;
    const int b    = blockIdx.x % BLOCKS_PER_SAMPLE;
    const int tid  = threadIdx.x;
    const int wave = tid >> 5;
    const int lane = tid & 31;
    const int l16  = lane & 15;
    const bool lo  = lane < 16;

    // tile id -> (i, jb, kb): D tile = S3[i, j0:j0+16, k0:k0+16]
    const int T  = b * 8 + wave;         // 0..431
    const int i  = T / 9;
    const int jb = (T % 9) / 3;
    const int kb = T % 3;
    const int j0 = jb * 16;
    const int k0 = kb * 16;

    const float* p = path + (size_t)n * LPTS * C;

    // producer: threads 0..191 build one (tt, j) increment of chunk c
    auto load_chunk = [&](int c, int buf) {
        if (tid < 4 * C) {
            int tt = tid / C;
            int j  = tid % C;
            int t  = c * 4 + tt;
            float v = 0.0f;
            if (t < TSTEPS) v = p[(t + 1) * C + j] - p[t * C + j];
            vbuf[buf][tt][j] = v;
            // speculative prefetch of rows two chunks ahead (L2 warm-up)
            int tp = t + 8;
            if (tp < LPTS) __builtin_prefetch(&p[tp * C + j], 0, 1);
        }
    };

    load_chunk(0, 0);
    __syncthreads();

    float s1i = 0.0f;        // S1 prefix, element i   (wave-uniform)
    float s2  = 0.0f;        // S2 prefix, row (i, j0 + lane%16), duplicated per half
    v8f acc = {0.f, 0.f, 0.f, 0.f, 0.f, 0.f, 0.f, 0.f};

    for (int c = 0; c < NCHUNK; ++c) {
        const int cb = c & 1;
        if (c + 1 < NCHUNK) load_chunk(c + 1, cb ^ 1);

        float m[4], kx[4];
        #pragma unroll
        for (int tt = 0; tt < 4; ++tt) {
            float vj = vbuf[cb][tt][j0 + l16];
            float vi = vbuf[cb][tt][i];      // wave-uniform LDS broadcast
            float vk = vbuf[cb][tt][k0 + l16];
            // M_t[i,j] = S2^{t-1}[i,j] + 0.5*S1^{t-1}[i]*v[j] + (1/6)*v[i]*v[j]
            m[tt]  = s2 + (0.5f * s1i + (1.0f / 6.0f) * vi) * vj;
            // S2 update: += (S1^{t-1}[i] + 0.5*v[i]) * v[j]
            s2    += (s1i + 0.5f * vi) * vj;
            s1i   += vi;
            kx[tt] = vk;
        }

        // A: 16x4 f32  (VGPR0 = K0 | K2 per lane half, VGPR1 = K1 | K3)
        // B: 4x16 f32  (mirrored layout)
        v2f A, B;
        A.x = lo ? m[0] : m[2];
        A.y = lo ? m[1] : m[3];
        B.x = lo ? kx[0] : kx[2];
        B.y = lo ? kx[1] : kx[3];

        acc = __builtin_amdgcn_wmma_f32_16x16x4_f32(
                  /*neg_a=*/false, A, /*neg_b=*/false, B,
                  /*c_mod=*/(short)0, acc, /*reuse_a=*/false, /*reuse_b=*/false);

        __syncthreads();
    }

    float* o = out + (size_t)n * OUT_STRIDE;

    // level 3: D tile -> S3[i, j, k].  D VGPR r: lanes 0-15 row r, lanes 16-31 row r+8
    const size_t base3 = S3_OFF + (size_t)i * C2;
    #pragma unroll
    for (int r = 0; r < 8; ++r) {
        int j = j0 + r + (lo ? 0 : 8);
        int k = k0 + l16;
        o[base3 + (size_t)j * C + k] = acc[r];
    }

    // level 2: each (i, jb) row slice tracked by 3 waves; kb==0 wave writes it
    if (kb == 0 && lo) {
        o[C + (size_t)i * C + j0 + l16] = s2;
    }

    // level 1: telescoped endpoint difference, one block per sample writes it
    if (b == 0 && tid < C) {
        o[tid] = p[(LPTS - 1) * C + tid] - p[tid];
    }
}

extern "C" void kernel_launch(void* const* d_in, const int* in_sizes, int n_in,
                              void* d_out, int out_size, void* d_ws, size_t ws_size,
                              hipStream_t stream)
{
    (void)in_sizes; (void)n_in; (void)out_size; (void)d_ws; (void)ws_size;
    const float* path = (const float*)d_in[0];
    float* out = (float*)d_out;
    dim3 grid(N_SAMPLES * BLOCKS_PER_SAMPLE);   // 1728 blocks
    sig3_fused_kernel<<<grid, 256, 0, stream>>>(path, out);
}